// InfoDrop_12756052869783
// MI455X (gfx1250) — compile-verified
//
#include <hip/hip_runtime.h>
#include <hip/hip_bf16.h>
#include <stdint.h>

// ---------------- problem constants (fixed by reference setup) ----------------
#define BB      16
#define CIN     3
#define OCH     64
#define HH      224
#define WW      224
#define HW      (HH*WW)            // 50176 = 256 * 196
#define SEG     196                // coarse-CDF segment length
#define NSEG    256                // HW / SEG
#define NSNV    9
#define KK      27                 // CIN*3*3
#define MROWS   (BB*HH*WW)         // 802816 (divisible by 16)
#define NDIFF   (BB*NSNV*HW)       // 7,225,344
#define COUNT   75264              // int(1.5*224*224)
#define NROWS   (BB*OCH)           // 1024
#define NSAMP   ((unsigned)NROWS*(unsigned)COUNT)  // 77,070,336 (even, %256==0)

// workspace layout (floats)
#define OFF_SUM   0                 // [0]  global diffs sum
#define OFF_BSUM  1                 // [1..16] per-batch pow-sums
#define OFF_DIFFS 64
#define OFF_T     (OFF_DIFFS + NDIFF)
#define OFF_CDF   (OFF_T + BB*HW)

typedef __attribute__((ext_vector_type(2))) float v2f;
typedef __attribute__((ext_vector_type(8))) float v8f;

// ---------------- helpers ----------------
__device__ __forceinline__ float ld_x(const float* __restrict__ X, int b, int c, int y, int x) {
    if ((unsigned)y >= (unsigned)HH || (unsigned)x >= (unsigned)WW) return 0.0f;
    return X[((b*CIN + c)*HH + y)*WW + x];
}
// implicit-GEMM A element: row = output pixel (b,y,x), col = k -> (c,kh,kw), pad=1
__device__ __forceinline__ float ldA(const float* __restrict__ X, int b, int y, int x, int k) {
    if (k >= KK) return 0.0f;
    int c = k / 9, r = (k % 9) / 3, s = k % 3;
    return ld_x(X, b, c, y + r - 1, x + s - 1);
}

// ---------------- 1) conv via fp32 WMMA implicit GEMM, fused bias+ReLU ----------------
// one wave per (16-row M tile, 16-col N tile); blockDim=128 -> 4 waves cover all 64 chans.
// 224%16==0 -> a tile never crosses an image row, so (b,y,xbase) are block-uniform (SALU).
__global__ void __launch_bounds__(128) conv_wmma_kernel(const float* __restrict__ X,
                                                        const float* __restrict__ Wt,
                                                        const float* __restrict__ Bs,
                                                        float* __restrict__ Y) {
    const int lane = threadIdx.x & 31;
    const int nt   = threadIdx.x >> 5;      // n-tile 0..3
    const int half = lane >> 4;             // hi/lo 16 lanes
    const int l16  = lane & 15;

    // block-uniform tile coordinates (scalar path)
    const int m0    = blockIdx.x * 16;
    const int xbase = m0 % WW;
    const int trow  = m0 / WW;
    const int y     = trow % HH;
    const int b     = trow / HH;

    const int x = xbase + l16;              // A row pixel for this lane

    v8f acc = {};
#if defined(__gfx1250__)
    for (int k0 = 0; k0 < 28; k0 += 4) {    // K=27 padded to 28 -> 7 WMMA steps
        const int ka = k0 + 2 * half;
        v2f av, bv;
        av.x = ldA(X, b, y, x, ka);
        av.y = ldA(X, b, y, x, ka + 1);
        const int n = nt * 16 + l16;        // B col for this lane
        bv.x = (ka     < KK) ? Wt[n * KK + ka]     : 0.0f;
        bv.y = (ka + 1 < KK) ? Wt[n * KK + ka + 1] : 0.0f;
        acc = __builtin_amdgcn_wmma_f32_16x16x4_f32(false, av, false, bv,
                                                    (short)0, acc, false, false);
    }
#endif
    const int n = nt * 16 + l16;
    const float bias = Bs[n];
    float* __restrict__ yrow = Y + (((size_t)b*OCH + n)*HH + y)*WW + xbase;
    #pragma unroll
    for (int v = 0; v < 8; ++v) {
        const int mrow = v + 8 * half;      // D row = pixel offset within tile
        float val = acc[v] + bias;
        yrow[mrow] = val > 0.0f ? val : 0.0f;
    }
}

// ---------------- 0) zero the atomic-reduction slots ----------------
__global__ void init_ws_kernel(float* wsf) {
    if (threadIdx.x < 32) wsf[threadIdx.x] = 0.0f;
}

// ---------------- 2) diffs = 3x3 box-sum of per-offset squared diffs; + global sum ----------------
__global__ void __launch_bounds__(256) diffs_kernel(const float* __restrict__ X,
                                                    const int* __restrict__ off_i,
                                                    const int* __restrict__ off_j,
                                                    float* __restrict__ wsf) {
    const int idx = blockIdx.x * 256 + threadIdx.x;   // (b,n,y,x)
    float local = 0.0f;
    if (idx < NDIFF) {
        const int xx = idx % WW;  int t = idx / WW;
        const int yy = t % HH;    t /= HH;
        const int n  = t % NSNV;  const int b = t / NSNV;
        const int oi = off_i[n] - 3;
        const int oj = off_j[n] - 3;
        float acc = 0.0f;
        #pragma unroll
        for (int wi = 0; wi < 3; ++wi) {
            #pragma unroll
            for (int wj = 0; wj < 3; ++wj) {
                const int iy = yy + wi - 1;           // origin X coord = i'-1, i'=yy+wi
                const int ix = xx + wj - 1;
                #pragma unroll
                for (int c = 0; c < CIN; ++c) {
                    const float o = ld_x(X, b, c, iy, ix);
                    const float s = ld_x(X, b, c, iy + oi, ix + oj);
                    const float d = s - o;
                    acc += d * d;
                }
            }
        }
        wsf[OFF_DIFFS + idx] = acc;
        local = acc;
    }
    __shared__ float red[256];
    red[threadIdx.x] = local; __syncthreads();
    for (int s = 128; s > 0; s >>= 1) {
        if (threadIdx.x < s) red[threadIdx.x] += red[threadIdx.x + s];
        __syncthreads();
    }
    if (threadIdx.x == 0) atomicAdd(&wsf[OFF_SUM], red[0]);
}

// ---------------- 3) t = (mean_n exp(-d/(2*mean)))^(1/T); + per-batch sums ----------------
__global__ void __launch_bounds__(256) prob_kernel(float* __restrict__ wsf) {
    const int idx = blockIdx.x * 256 + threadIdx.x;   // (b, pos)
    const float mean = wsf[OFF_SUM] / (float)NDIFF;
    const float inv2m = 1.0f / (2.0f * mean);
    const int b = idx / HW;
    const int pos = idx % HW;
    float s = 0.0f;
    #pragma unroll
    for (int n = 0; n < NSNV; ++n)
        s += __expf(-wsf[OFF_DIFFS + ((b*NSNV + n)*HW + pos)] * inv2m);
    s *= (1.0f / (float)NSNV);
    const float tv = __powf(s, 1.0f / 0.03f);
    wsf[OFF_T + idx] = tv;
    __shared__ float red[256];
    red[threadIdx.x] = tv; __syncthreads();
    for (int st = 128; st > 0; st >>= 1) {
        if (threadIdx.x < st) red[threadIdx.x] += red[threadIdx.x + st];
        __syncthreads();
    }
    // HW % 256 == 0 -> every block lies inside one batch
    if (threadIdx.x == 0) atomicAdd(&wsf[OFF_BSUM + b], red[0]);
}

// ---------------- 4) per-batch inclusive scan -> unnormalized CDF ----------------
__global__ void __launch_bounds__(256) cdf_kernel(float* __restrict__ wsf) {
    const int b = blockIdx.x;
    const int tid = threadIdx.x;
    const float inv = 1.0f / wsf[OFF_BSUM + b];
    __shared__ float sd[256];
    __shared__ float carry;
    if (tid == 0) carry = 0.0f;
    __syncthreads();
    for (int base = 0; base < HW; base += 256) {      // HW/256 = 196 chunks
        const int i = base + tid;
        float v = wsf[OFF_T + b*HW + i] * inv + 1e-8f;
        sd[tid] = v; __syncthreads();
        for (int off = 1; off < 256; off <<= 1) {
            float add = (tid >= off) ? sd[tid - off] : 0.0f;
            __syncthreads();
            sd[tid] += add;
            __syncthreads();
        }
        const float res = sd[tid] + carry;
        wsf[OFF_CDF + b*HW + i] = res;
        __syncthreads();
        if (tid == 255) carry = res;
        __syncthreads();
    }
}

// ---------------- 5) threefry2x32 draws -> two-level inverse-CDF -> scatter zeros ----------------
struct U2 { unsigned a, b; };
__device__ __forceinline__ U2 threefry2x32(unsigned k0, unsigned k1, unsigned x0, unsigned x1) {
    const unsigned ks2 = k0 ^ k1 ^ 0x1BD11BDAu;
    x0 += k0; x1 += k1;
    const unsigned R0[4] = {13u, 15u, 26u, 6u};
    const unsigned R1[4] = {17u, 29u, 16u, 24u};
    #pragma unroll
    for (int i = 0; i < 4; ++i) { x0 += x1; x1 = (x1 << R0[i]) | (x1 >> (32 - R0[i])); x1 ^= x0; }
    x0 += k1;  x1 += ks2 + 1u;
    #pragma unroll
    for (int i = 0; i < 4; ++i) { x0 += x1; x1 = (x1 << R1[i]) | (x1 >> (32 - R1[i])); x1 ^= x0; }
    x0 += ks2; x1 += k0 + 2u;
    #pragma unroll
    for (int i = 0; i < 4; ++i) { x0 += x1; x1 = (x1 << R0[i]) | (x1 >> (32 - R0[i])); x1 ^= x0; }
    x0 += k0;  x1 += k1 + 3u;
    #pragma unroll
    for (int i = 0; i < 4; ++i) { x0 += x1; x1 = (x1 << R1[i]) | (x1 >> (32 - R1[i])); x1 ^= x0; }
    x0 += k1;  x1 += ks2 + 4u;
    #pragma unroll
    for (int i = 0; i < 4; ++i) { x0 += x1; x1 = (x1 << R0[i]) | (x1 >> (32 - R0[i])); x1 ^= x0; }
    x0 += ks2; x1 += k0 + 5u;
    U2 r; r.a = x0; r.b = x1; return r;
}

// COUNT*64 = 4,816,896 is divisible by 256, so every 256-thread block shares one batch b:
// stage a 256-entry coarse CDF (segment maxima) in LDS, search 8 steps there, then
// only ~8 dependent global steps inside one 196-element segment.
__global__ void __launch_bounds__(256) mask_kernel(const float* __restrict__ wsf,
                                                   float* __restrict__ out) {
    const unsigned g = blockIdx.x * 256u + threadIdx.x;      // < NSAMP
    const unsigned row = g / (unsigned)COUNT;                // (b*64 + o), uniform-ish
    const unsigned b   = row >> 6;                           // block-uniform (see above)

    const float* __restrict__ c = wsf + OFF_CDF + b * HW;

    __shared__ float coarse[NSEG];
    coarse[threadIdx.x] = c[threadIdx.x * SEG + (SEG - 1)];  // max of segment i
    __syncthreads();

    // jax.random.uniform(key(42), (B*O, COUNT)) bit-stream
    const unsigned HALF = NSAMP / 2u;
    unsigned x0, x1;
    const bool lo_half = g < HALF;
    if (lo_half) { x0 = g;        x1 = g + HALF; }
    else         { x0 = g - HALF; x1 = g;        }
    const U2 r = threefry2x32(0u, 42u, x0, x1);
    const unsigned bits = lo_half ? r.a : r.b;
    const float u = __uint_as_float((bits >> 9) | 0x3f800000u) - 1.0f;
    const float target = u * c[HW - 1];                      // fold CDF normalization into u

    // level 1: find first segment whose max >= target (LDS, 8 steps)
    int slo = 0, shi = NSEG - 1;                             // last segment always >= target
    while (slo < shi) {
        const int mid = (slo + shi) >> 1;
        if (coarse[mid] < target) slo = mid + 1; else shi = mid;
    }
    // level 2: searchsorted-left within segment (global, 8 steps)
    int lo = slo * SEG, hi = lo + SEG;
    while (lo < hi) {
        const int mid = (lo + hi) >> 1;
        if (c[mid] < target) lo = mid + 1; else hi = mid;
    }
    if (lo > HW - 1) lo = HW - 1;
    out[row * (unsigned)HW + lo] = 0.0f;                     // y * mask (idempotent scatter)
}

// ---------------- launcher ----------------
extern "C" void kernel_launch(void* const* d_in, const int* in_sizes, int n_in,
                              void* d_out, int out_size, void* d_ws, size_t ws_size,
                              hipStream_t stream) {
    const float* x      = (const float*)d_in[0];
    const float* conv_w = (const float*)d_in[1];
    const float* conv_b = (const float*)d_in[2];
    const int*   off_i  = (const int*)d_in[3];
    const int*   off_j  = (const int*)d_in[4];
    float* out = (float*)d_out;
    float* wsf = (float*)d_ws;

    init_ws_kernel<<<1, 32, 0, stream>>>(wsf);

    conv_wmma_kernel<<<MROWS / 16, 128, 0, stream>>>(x, conv_w, conv_b, out);

    diffs_kernel<<<(NDIFF + 255) / 256, 256, 0, stream>>>(x, off_i, off_j, wsf);

    prob_kernel<<<(BB * HW) / 256, 256, 0, stream>>>(wsf);

    cdf_kernel<<<BB, 256, 0, stream>>>(wsf);

    mask_kernel<<<NSAMP / 256u, 256, 0, stream>>>(wsf, out);
}